// GraphStructureLoss_87660282511507
// MI455X (gfx1250) — compile-verified
//
#include <hip/hip_runtime.h>
#include <hip/hip_bf16.h>

#define N     8192
#define TILE  64
#define TPAD  65           // LDS row stride (floats) to de-conflict banks
#define NT    (N / TILE)   // 128 tiles per dim

typedef __attribute__((ext_vector_type(2))) float v2f;
typedef __attribute__((ext_vector_type(8))) float v8f;

// ---------------------------------------------------------------------------
// Kernel 1: s[i] = rsqrt( sum_j relu(adj[i,j]) + 1e-10 )
// One block per row, fully coalesced float4 reads.
// ---------------------------------------------------------------------------
__global__ void gsl_rowsum_kernel(const float* __restrict__ adj,
                                  float* __restrict__ s) {
    __shared__ float red[256];
    const int row = blockIdx.x;
    const float4* rp = (const float4*)(adj + (size_t)row * N);
    float acc = 0.f;
#pragma unroll
    for (int k = 0; k < N / 4; k += 256) {
        float4 v = rp[k + threadIdx.x];
        acc += fmaxf(v.x, 0.f) + fmaxf(v.y, 0.f) + fmaxf(v.z, 0.f) + fmaxf(v.w, 0.f);
    }
    red[threadIdx.x] = acc;
    __syncthreads();
    for (int off = 128; off > 0; off >>= 1) {
        if (threadIdx.x < (unsigned)off) red[threadIdx.x] += red[threadIdx.x + off];
        __syncthreads();
    }
    if (threadIdx.x == 0) s[row] = rsqrtf(red[0] + 1e-10f);
}

// ---------------------------------------------------------------------------
// Kernel 2: one block per unordered tile pair (bi <= bj).
//   Loads tile P=adj[bi,bj] and Q=adj[bj,bi] into LDS (each element of adj
//   is read exactly once across the whole grid).
//   Per element: a=A_ij, b=A_ji, t=a+b -> rc row/col sums; sumA; sum (a-b)^2.
//   Row/col sums of the 64x64 t-matrix are reduced with
//   V_WMMA_F32_16X16X4_F32 against an all-ones B operand.
// ---------------------------------------------------------------------------
__global__ void gsl_pair_kernel(const float* __restrict__ adj,
                                const float* __restrict__ s,
                                float* __restrict__ rc,
                                float* __restrict__ acc) {
    __shared__ float P[TILE * TPAD];
    __shared__ float Q[TILE * TPAD];
    __shared__ float T[TILE * TPAD];
    __shared__ float sr[TILE];
    __shared__ float scv[TILE];
    __shared__ float red[2 * TILE];   // [0,64): row sums  [64,128): col sums
    __shared__ float sa[256];
    __shared__ float sb[256];

    const int tid = threadIdx.x;

    // ---- decode blockIdx.x -> (bi, bj), bi <= bj  (triangular enumeration)
    const int p = blockIdx.x;
    int bi = (int)((2.0f * NT + 1.0f -
                    sqrtf((float)(2 * NT + 1) * (2 * NT + 1) - 8.0f * (float)p)) * 0.5f);
    if (bi < 0) bi = 0;
    if (bi > NT - 1) bi = NT - 1;
    while (bi > 0 && (bi * NT - (bi * (bi - 1)) / 2) > p) --bi;
    while (((bi + 1) * NT - ((bi + 1) * bi) / 2) <= p) ++bi;
    const int bj = bi + (p - (bi * NT - (bi * (bi - 1)) / 2));
    const bool diag = (bi == bj);

    // ---- stage 1: cooperative coalesced tile loads into LDS
    {
        const float* Pg = adj + (size_t)bi * TILE * N + (size_t)bj * TILE;
#pragma unroll
        for (int k = 0; k < 4; ++k) {
            int g  = tid + k * 256;          // 1024 float4s = 64 rows x 16
            int r  = g >> 4;
            int c4 = (g & 15) << 2;
            float4 v = *(const float4*)(Pg + (size_t)r * N + c4);
            P[r * TPAD + c4 + 0] = v.x;
            P[r * TPAD + c4 + 1] = v.y;
            P[r * TPAD + c4 + 2] = v.z;
            P[r * TPAD + c4 + 3] = v.w;
        }
        if (!diag) {
            const float* Qg = adj + (size_t)bj * TILE * N + (size_t)bi * TILE;
#pragma unroll
            for (int k = 0; k < 4; ++k) {
                int g  = tid + k * 256;
                int r  = g >> 4;
                int c4 = (g & 15) << 2;
                float4 v = *(const float4*)(Qg + (size_t)r * N + c4);
                Q[r * TPAD + c4 + 0] = v.x;
                Q[r * TPAD + c4 + 1] = v.y;
                Q[r * TPAD + c4 + 2] = v.z;
                Q[r * TPAD + c4 + 3] = v.w;
            }
        }
        if (tid < TILE)                      sr[tid]         = s[bi * TILE + tid];
        else if (tid < 2 * TILE)             scv[tid - TILE] = s[bj * TILE + (tid - TILE)];
    }
    __syncthreads();

    const float* Qp = diag ? P : Q;

    // ---- stage 2: elementwise contributions + t-matrix into LDS
    {
        const int u = tid >> 2;              // 0..63  (row in bi tile)
        const int q = tid & 3;               // column quarter
        const float su = sr[u];
        float sumA = 0.f, sumSq = 0.f;
#pragma unroll
        for (int i = 0; i < 16; ++i) {
            int v = (q << 4) + i;
            float x  = fmaxf(P[u * TPAD + v], 0.f);    // B_ij
            float y  = fmaxf(Qp[v * TPAD + u], 0.f);   // B_ji
            float sc = su * scv[v];
            float a  = x * sc;                         // A_ij
            float b  = y * sc;                         // A_ji
            float d  = a - b;
            T[u * TPAD + v] = a + b;
            if (diag) { sumA += a;     sumSq += d * d; }
            else      { sumA += a + b; sumSq += 2.f * d * d; }
        }
        sa[tid] = sumA;
        sb[tid] = sumSq;
    }
    __syncthreads();

    // ---- stage 3: WMMA reductions of T (B = ones so D[m][*] = row sums)
    {
        const int wave = tid >> 5;
        const int lane = tid & 31;
        const int m    = lane & 15;
        const int koff = (lane >> 4) << 1;   // A 16x4 f32 layout: lanes 16-31 hold K+2
        v8f c = {};
        v2f bones = {1.0f, 1.0f};
        if (wave < 4) {                      // row sums, rows [16w, 16w+16)
            const int rbase = wave << 4;
            const int rowm  = rbase + m;
            for (int k0 = 0; k0 < TILE; k0 += 4) {
                v2f a;
                a.x = T[rowm * TPAD + k0 + koff];
                a.y = T[rowm * TPAD + k0 + koff + 1];
                c = __builtin_amdgcn_wmma_f32_16x16x4_f32(
                        false, a, false, bones, (short)0, c, false, false);
            }
            if (lane == 0) {
#pragma unroll
                for (int i = 0; i < 8; ++i) red[rbase + i] = c[i];
            }
            if (lane == 16) {
#pragma unroll
                for (int i = 0; i < 8; ++i) red[rbase + 8 + i] = c[i];
            }
        } else {                             // col sums via transposed A reads
            const int cbase = (wave - 4) << 4;
            const int colm  = cbase + m;
            for (int k0 = 0; k0 < TILE; k0 += 4) {
                v2f a;
                a.x = T[(k0 + koff) * TPAD + colm];
                a.y = T[(k0 + koff + 1) * TPAD + colm];
                c = __builtin_amdgcn_wmma_f32_16x16x4_f32(
                        false, a, false, bones, (short)0, c, false, false);
            }
            if (lane == 0) {
#pragma unroll
                for (int i = 0; i < 8; ++i) red[TILE + cbase + i] = c[i];
            }
            if (lane == 16) {
#pragma unroll
                for (int i = 0; i < 8; ++i) red[TILE + cbase + 8 + i] = c[i];
            }
        }
    }
    __syncthreads();

    // ---- stage 4: global accumulation
    if (tid < TILE) {
        atomicAdd(&rc[bi * TILE + tid], red[tid]);
    } else if (!diag && tid < 2 * TILE) {
        atomicAdd(&rc[bj * TILE + (tid - TILE)], red[tid]);
    }

    if (tid < 128) { sa[tid] += sa[tid + 128]; sb[tid] += sb[tid + 128]; }
    __syncthreads();
    if (tid < 64)  { sa[tid] += sa[tid + 64];  sb[tid] += sb[tid + 64];  }
    __syncthreads();
    if (tid < 32)  { sa[tid] += sa[tid + 32];  sb[tid] += sb[tid + 32];  }
    __syncthreads();
    if (tid < 16)  { sa[tid] += sa[tid + 16];  sb[tid] += sb[tid + 16];  }
    __syncthreads();
    if (tid < 8)   { sa[tid] += sa[tid + 8];   sb[tid] += sb[tid + 8];   }
    __syncthreads();
    if (tid < 4)   { sa[tid] += sa[tid + 4];   sb[tid] += sb[tid + 4];   }
    __syncthreads();
    if (tid == 0) {
        atomicAdd(&acc[0], sa[0] + sa[1] + sa[2] + sa[3]);   // sum A
        atomicAdd(&acc[1], sb[0] + sb[1] + sb[2] + sb[3]);   // sum (A-A^T)^2
    }
}

// ---------------------------------------------------------------------------
// Kernel 3: loss = smooth + sqrt(sumSq) + sumA
//   smooth = sum_i (0.5*rc[i] - A_ii) / (0.5*rc[i] + 1e-3)
// ---------------------------------------------------------------------------
__global__ void gsl_final_kernel(const float* __restrict__ adj,
                                 const float* __restrict__ s,
                                 const float* __restrict__ rc,
                                 const float* __restrict__ acc,
                                 float* __restrict__ out) {
    __shared__ float red[256];
    float sm = 0.f;
    for (int i = threadIdx.x; i < N; i += 256) {
        float diagA = fmaxf(adj[(size_t)i * N + i], 0.f) * s[i] * s[i];
        float rs = 0.5f * rc[i];
        sm += (rs - diagA) / (rs + 1e-3f);
    }
    red[threadIdx.x] = sm;
    __syncthreads();
    for (int off = 128; off > 0; off >>= 1) {
        if (threadIdx.x < (unsigned)off) red[threadIdx.x] += red[threadIdx.x + off];
        __syncthreads();
    }
    if (threadIdx.x == 0) out[0] = red[0] + sqrtf(acc[1]) + acc[0];
}

// ---------------------------------------------------------------------------
extern "C" void kernel_launch(void* const* d_in, const int* in_sizes, int n_in,
                              void* d_out, int out_size, void* d_ws, size_t ws_size,
                              hipStream_t stream) {
    const float* adj = (const float*)d_in[0];
    float* s   = (float*)d_ws;          // [0, N)       rsqrt degrees
    float* rc  = s + N;                 // [N, 2N)      row+col sums of A
    float* acc = rc + N;                // [2N, 2N+4)   {sumA, sumSq, -, -}

    // zero rc + acc (contiguous); s is fully overwritten by kernel 1
    hipMemsetAsync(rc, 0, (size_t)(N + 4) * sizeof(float), stream);

    gsl_rowsum_kernel<<<N, 256, 0, stream>>>(adj, s);

    const int pairs = NT * (NT + 1) / 2;   // 8256 tile-pair blocks
    gsl_pair_kernel<<<pairs, 256, 0, stream>>>(adj, s, rc, acc);

    gsl_final_kernel<<<1, 256, 0, stream>>>(adj, s, rc, acc, (float*)d_out);
}